// Full_Graph_45612552684092
// MI455X (gfx1250) — compile-verified
//
#include <hip/hip_runtime.h>
#include <hip/hip_bf16.h>

typedef float v2f __attribute__((ext_vector_type(2)));
typedef float v8f __attribute__((ext_vector_type(8)));

#define CC    10
#define C2    20
#define HW    16384
#define NPIX  131072
#define NPAIR 210

// ws layout (floats). A1/c1 padded to 32 M-rows, A2/c2 padded to 16, so the
// per-pixel WMMA kernels are completely branch-free (padded rows are zero).
#define OFF_S1U 0
#define OFF_S2U 20
#define OFF_S1X 230
#define OFF_S2X 250
#define OFF_A1  460      // 32*20
#define OFF_C1  1100     // 32
#define OFF_A2  1132     // 16*20
#define OFF_C2V 1452     // 16

__device__ __forceinline__ float sigmoidf(float x) { return 1.f / (1.f + __expf(-x)); }

// Block-level moment accumulation: 256 pixels staged in LDS rows [C2][257].
// Threads 0..19 reduce means, threads 20..229 reduce the 210 unique u_i*u_j pairs.
__device__ __forceinline__ void accumulate_moments(const float (*m)[257],
                                                   float* __restrict__ S1,
                                                   float* __restrict__ S2, int t) {
  if (t < C2) {
    float s = 0.f;
    for (int q = 0; q < 256; ++q) s += m[t][q];
    atomicAdd(&S1[t], s);
  } else if (t < C2 + NPAIR) {
    int pidx = t - C2;
    int i = 0, r = pidx;
    while (r >= C2 - i) { r -= C2 - i; ++i; }
    int j = i + r;
    float s = 0.f;
    for (int q = 0; q < 256; ++q) s += m[i][q] * m[j][q];
    atomicAdd(&S2[pidx], s);
  }
}

// outS[m][n] = relu( sum_k A[m][k]*inS[k][n] + cv[m] ) for m in [0, NMT*16),
// k in [0,20), n over the block's 256 pixels. A is padded to NMT*16 rows
// (zeros beyond the real output count), so everything is unconditional.
// V_WMMA_F32_16X16X4_F32: 8 waves x 2 N-tiles(16 px) x NMT M-tiles x 5 K-chunks.
template <int NMT>
__device__ __forceinline__ void wmma_linear_relu(const float (*inS)[257], float (*outS)[257],
                                                 const float* __restrict__ A,
                                                 const float* __restrict__ cv, int t) {
  const int lane = t & 31;
  const int wave = t >> 5;
  const int hi = lane >> 4;          // lanes 16..31
  const int ln = lane & 15;
  const int koff0 = hi ? 2 : 0;      // A/B fragment K placement (ISA 7.12.2)
  const int koff1 = hi ? 3 : 1;

  // Hoist A fragments + bias accumulator init (loop-invariant across N-tiles).
  v2f afrag[NMT][5];
  v8f cinit[NMT];
  for (int mt = 0; mt < NMT; ++mt) {
    const int row = mt * 16 + ln;
    for (int kc = 0; kc < 5; ++kc) {
      afrag[mt][kc][0] = A[row * C2 + kc * 4 + koff0];
      afrag[mt][kc][1] = A[row * C2 + kc * 4 + koff1];
    }
    for (int v = 0; v < 8; ++v) cinit[mt][v] = cv[mt * 16 + (hi ? 8 : 0) + v];
  }

  for (int tile = 2 * wave; tile < 2 * wave + 2; ++tile) {
    const int n = tile * 16 + ln;    // pixel within block
    for (int mt = 0; mt < NMT; ++mt) {
      v8f acc = cinit[mt];
      for (int kc = 0; kc < 5; ++kc) {           // K = 20 = 5 x 4
        v2f bv;
        bv[0] = inS[kc * 4 + koff0][n];
        bv[1] = inS[kc * 4 + koff1][n];
        acc = __builtin_amdgcn_wmma_f32_16x16x4_f32(false, afrag[mt][kc], false, bv,
                                                    (short)0, acc, false, false);
      }
      for (int v = 0; v < 8; ++v)                // D: VGPR v -> M = v (+8 for hi lanes)
        outS[mt * 16 + (hi ? 8 : 0) + v][n] = fmaxf(acc[v], 0.f);
    }
  }
}

__global__ void k_init(float* __restrict__ ws) {
  int t = blockIdx.x * blockDim.x + threadIdx.x;
  if (t < 460) ws[t] = 0.f;   // zero S1u,S2u,S1x,S2x
}

__global__ __launch_bounds__(256) void k_pass1(
    const float* __restrict__ f0, const float* __restrict__ f1,
    const float* __restrict__ h0, const float* __restrict__ h1, const float* __restrict__ h2,
    const float* __restrict__ Watt, const float* __restrict__ batt,
    const float* __restrict__ Wg0, const float* __restrict__ bg0, const float* __restrict__ Wc0,
    float* __restrict__ node0, float* __restrict__ att, float* __restrict__ ws) {
  __shared__ float uS[C2][257];
  const int t = threadIdx.x;
  const int p = blockIdx.x * 256 + t;
  const int b = p >> 14, hw = p & (HW - 1);
  const size_t base = (size_t)b * CC * HW + hw;
  float f0v[CC], f1v[CC], h0v[CC], h1v[CC], h2v[CC];
  for (int c = 0; c < CC; ++c) {
    size_t o = base + (size_t)c * HW;
    f0v[c] = f0[o]; f1v[c] = f1[o]; h0v[c] = h0[o]; h1v[c] = h1[o]; h2v[c] = h2[o];
  }
  // comp_att = sigmoid(W_att . [h1,h2] + b_att)
  float a = batt[0];
  for (int c = 0; c < CC; ++c) a += Watt[c] * h1v[c] + Watt[CC + c] * h2v[c];
  a = sigmoidf(a);
  att[p] = a;
  // node0 = ConvGRU(x=h0, h=f0)
  float g0 = bg0[0], g1 = bg0[1];
  for (int c = 0; c < CC; ++c) {
    g0 += Wg0[c] * h0v[c] + Wg0[CC + c] * f0v[c];
    g1 += Wg0[C2 + c] * h0v[c] + Wg0[C2 + CC + c] * f0v[c];
  }
  const float rst = sigmoidf(g0), upd = sigmoidf(g1);
  for (int o = 0; o < CC; ++o) {
    float s = 0.f;
    for (int c = 0; c < CC; ++c)
      s += Wc0[o * C2 + c] * h0v[c] + Wc0[o * C2 + CC + c] * (rst * f0v[c]);
    node0[base + (size_t)o * HW] = (1.f - upd) * f0v[o] + upd * tanhf(s);
  }
  // u = [f1, (h1+h2)*att] -> LDS, then block moment reduction
  for (int c = 0; c < CC; ++c) {
    uS[c][t] = f1v[c];
    uS[CC + c][t] = (h1v[c] + h2v[c]) * a;
  }
  __syncthreads();
  accumulate_moments(uS, ws + OFF_S1U, ws + OFF_S2U, t);
}

// Fold training-mode BN into affine: A = diag(gamma*rsqrt(var+eps)) * W ;
// c = beta - gamma*rsqrt*mean, with mean_c = w_c.mu, var_c = w_c^T Cov w_c
// (biased, ddof=0). Rows [nout, npad) are written as zeros (WMMA padding).
__global__ void k_finalize(const float* __restrict__ S1, const float* __restrict__ S2,
                           const float* __restrict__ W, const float* __restrict__ gam,
                           const float* __restrict__ bet,
                           float* __restrict__ A, float* __restrict__ cvec,
                           int nout, int npad) {
  const int t = threadIdx.x;
  if (t >= npad) return;
  if (t >= nout) {
    for (int i = 0; i < C2; ++i) A[t * C2 + i] = 0.f;
    cvec[t] = 0.f;
    return;
  }
  const float invN = 1.f / (float)NPIX;
  float mu[C2];
  for (int i = 0; i < C2; ++i) mu[i] = S1[i] * invN;
  float m = 0.f;
  for (int i = 0; i < C2; ++i) m += W[t * C2 + i] * mu[i];
  float var = 0.f;
  for (int i = 0; i < C2; ++i)
    for (int j = 0; j < C2; ++j) {
      const int lo = i < j ? i : j, hi = i < j ? j : i;
      const int pidx = lo * C2 - (lo * (lo - 1)) / 2 + (hi - lo);
      const float cov = S2[pidx] * invN - mu[i] * mu[j];
      var += W[t * C2 + i] * W[t * C2 + j] * cov;
    }
  const float sc = gam[t] * rsqrtf(var + 1e-5f);
  for (int i = 0; i < C2; ++i) A[t * C2 + i] = sc * W[t * C2 + i];
  cvec[t] = bet[t] - sc * m;
}

__global__ __launch_bounds__(256) void k_pass3(
    const float* __restrict__ f1, const float* __restrict__ h1, const float* __restrict__ h2,
    const float* __restrict__ att, float* __restrict__ ws) {
  __shared__ float uS[C2][257];
  __shared__ float xS[32][257];
  const int t = threadIdx.x;
  const int p = blockIdx.x * 256 + t;
  const int b = p >> 14, hw = p & (HW - 1);
  const size_t base = (size_t)b * CC * HW + hw;
  const float a = att[p];
  for (int c = 0; c < CC; ++c) {
    size_t o = base + (size_t)c * HW;
    uS[c][t] = f1[o];
    uS[CC + c][t] = (h1[o] + h2[o]) * a;
  }
  __syncthreads();
  wmma_linear_relu<2>(uS, xS, ws + OFF_A1, ws + OFF_C1, t);
  __syncthreads();
  accumulate_moments((const float (*)[257])xS, ws + OFF_S1X, ws + OFF_S2X, t);
}

__global__ __launch_bounds__(256) void k_pass5(
    const float* __restrict__ f1, const float* __restrict__ h1, const float* __restrict__ h2,
    const float* __restrict__ att, const float* __restrict__ ws,
    const float* __restrict__ Wg1, const float* __restrict__ bg1, const float* __restrict__ Wc1,
    float* __restrict__ node1) {
  __shared__ float bufA[C2][257];   // uS, then reused (after barrier) as comp_full (16 rows)
  __shared__ float xS[32][257];
  const int t = threadIdx.x;
  const int p = blockIdx.x * 256 + t;
  const int b = p >> 14, hw = p & (HW - 1);
  const size_t base = (size_t)b * CC * HW + hw;
  float f1v[CC];
  const float a = att[p];
  for (int c = 0; c < CC; ++c) {
    size_t o = base + (size_t)c * HW;
    f1v[c] = f1[o];
    bufA[c][t] = f1v[c];
    bufA[CC + c][t] = (h1[o] + h2[o]) * a;
  }
  __syncthreads();
  wmma_linear_relu<2>(bufA, xS, ws + OFF_A1, ws + OFF_C1, t);   // x = relu(BN1(conv_r1(u)))
  __syncthreads();                                              // u dead; bufA reused below
  wmma_linear_relu<1>((const float (*)[257])xS, bufA,
                      ws + OFF_A2, ws + OFF_C2V, t);            // comp_full = relu(BN2(conv_r2(x)))
  __syncthreads();
  float cf[CC];
  for (int c = 0; c < CC; ++c) cf[c] = bufA[c][t];
  // node1 = ConvGRU(x=comp_full, h=f1)
  float g0 = bg1[0], g1 = bg1[1];
  for (int c = 0; c < CC; ++c) {
    g0 += Wg1[c] * cf[c] + Wg1[CC + c] * f1v[c];
    g1 += Wg1[C2 + c] * cf[c] + Wg1[C2 + CC + c] * f1v[c];
  }
  const float rst = sigmoidf(g0), upd = sigmoidf(g1);
  for (int o = 0; o < CC; ++o) {
    float s = 0.f;
    for (int c = 0; c < CC; ++c)
      s += Wc1[o * C2 + c] * cf[c] + Wc1[o * C2 + CC + c] * (rst * f1v[c]);
    node1[base + (size_t)o * HW] = (1.f - upd) * f1v[o] + upd * tanhf(s);
  }
}

extern "C" void kernel_launch(void* const* d_in, const int* in_sizes, int n_in,
                              void* d_out, int out_size, void* d_ws, size_t ws_size,
                              hipStream_t stream) {
  const float* f0    = (const float*)d_in[0];
  const float* f1    = (const float*)d_in[1];
  const float* h0    = (const float*)d_in[2];
  const float* h1    = (const float*)d_in[3];
  const float* h2    = (const float*)d_in[4];
  const float* W_att = (const float*)d_in[5];
  const float* b_att = (const float*)d_in[6];
  const float* W_r1  = (const float*)d_in[7];
  const float* g_r1  = (const float*)d_in[8];
  const float* be_r1 = (const float*)d_in[9];
  const float* W_r2  = (const float*)d_in[10];
  const float* g_r2  = (const float*)d_in[11];
  const float* be_r2 = (const float*)d_in[12];
  const float* Wg0   = (const float*)d_in[13];
  const float* bg0   = (const float*)d_in[14];
  const float* Wc0   = (const float*)d_in[15];
  const float* Wg1   = (const float*)d_in[16];
  const float* bg1   = (const float*)d_in[17];
  const float* Wc1   = (const float*)d_in[18];

  float* out   = (float*)d_out;
  float* node0 = out;                         // [8,10,128,128]
  float* node1 = out + (size_t)NPIX * CC;     // [8,10,128,128]
  float* att   = out + (size_t)NPIX * CC * 2; // [8,1,128,128]
  float* ws    = (float*)d_ws;

  const int nblk = NPIX / 256; // 512

  k_init<<<2, 256, 0, stream>>>(ws);
  k_pass1<<<nblk, 256, 0, stream>>>(f0, f1, h0, h1, h2, W_att, b_att,
                                    Wg0, bg0, Wc0, node0, att, ws);
  k_finalize<<<1, 32, 0, stream>>>(ws + OFF_S1U, ws + OFF_S2U, W_r1, g_r1, be_r1,
                                   ws + OFF_A1, ws + OFF_C1, C2, 32);
  k_pass3<<<nblk, 256, 0, stream>>>(f1, h1, h2, att, ws);
  k_finalize<<<1, 32, 0, stream>>>(ws + OFF_S1X, ws + OFF_S2X, W_r2, g_r2, be_r2,
                                   ws + OFF_A2, ws + OFF_C2V, CC, 16);
  k_pass5<<<nblk, 256, 0, stream>>>(f1, h1, h2, att, ws, Wg1, bg1, Wc1, node1);
}